// GATLayer_72516227826097
// MI455X (gfx1250) — compile-verified
//
#include <hip/hip_runtime.h>
#include <math.h>

#define N_NODES 50000
#define N_EDGES 800000
#define IN_DIM  256
#define OUT_DIM 128
#define LEAK    0.2f

typedef __attribute__((ext_vector_type(2))) float v2f;
typedef __attribute__((ext_vector_type(4))) float v4f;
typedef __attribute__((ext_vector_type(8))) float v8f;

// ---------------------------------------------------------------------------
// Kernel 0: zero the output accumulator + per-node reduction state
// ---------------------------------------------------------------------------
__global__ __launch_bounds__(256) void k_init(float* __restrict__ out,
                                              float* __restrict__ denom,
                                              unsigned* __restrict__ m_enc) {
  const int i = blockIdx.x * blockDim.x + threadIdx.x;
  if (i < N_NODES * OUT_DIM) out[i] = 0.0f;
  if (i < N_NODES) { denom[i] = 0.0f; m_enc[i] = 0u; }  // enc(0)=below all reals
}

// ---------------------------------------------------------------------------
// Kernel 1: z = h @ W via V_WMMA_F32_16X16X4_F32.
// Block = 256 threads = 8 waves; wave w owns N-tile [16w,16w+16); block owns
// M-tile [16*blockIdx.x, ...). 50000 % 16 == 0, so no edge guards.
// A 16x4 f32 frag: lane<16 -> {K0,K1}, lane>=16 -> {K2,K3} of row M=lane%16.
// B 4x16  f32 frag: lane<16 -> rows {K0,K1}, lane>=16 -> {K2,K3}, col N=lane%16.
// C/D: VGPR i holds row i (lanes 0-15) / row 8+i (lanes 16-31), col = lane%16.
// ---------------------------------------------------------------------------
__global__ __launch_bounds__(256) void k_gemm(const float* __restrict__ h,
                                              const float* __restrict__ W,
                                              float* __restrict__ z) {
  const int lane = threadIdx.x & 31;
  const int wv   = threadIdx.x >> 5;   // 0..7 -> N tile
  const int half = lane >> 4;          // 0 or 1
  const int l    = lane & 15;
  const int m0   = blockIdx.x * 16;
  const int n0   = wv * 16;

  const float* __restrict__ ar = h + (size_t)(m0 + l) * IN_DIM;  // A row
  const float* __restrict__ bc = W + (size_t)(n0 + l);           // B column

  v8f acc = {};
#pragma unroll 4
  for (int k = 0; k < IN_DIM; k += 4) {
    const int kk = k + 2 * half;
    v2f a; a.x = ar[kk];                     a.y = ar[kk + 1];
    v2f b; b.x = bc[(size_t)kk * OUT_DIM];   b.y = bc[(size_t)(kk + 1) * OUT_DIM];
    acc = __builtin_amdgcn_wmma_f32_16x16x4_f32(
        /*neg_a=*/false, a, /*neg_b=*/false, b,
        /*c_mod=*/(short)0, acc, /*reuse_a=*/false, /*reuse_b=*/false);
  }

  float* zp = z + (size_t)(m0 + 8 * half) * OUT_DIM + n0 + l;
#pragma unroll
  for (int i = 0; i < 8; ++i) zp[(size_t)i * OUT_DIM] = acc[i];
}

// ---------------------------------------------------------------------------
// Kernel 2: per-node attention scores  s1[n]=z[n].a1, s2[n]=z[n].a2
// ---------------------------------------------------------------------------
__global__ __launch_bounds__(256) void k_scores(const float* __restrict__ z,
                                                const float* __restrict__ attn_w,
                                                float* __restrict__ s1,
                                                float* __restrict__ s2) {
  const int n = blockIdx.x * blockDim.x + threadIdx.x;
  if (n >= N_NODES) return;
  const float* zr = z + (size_t)n * OUT_DIM;
  float d1 = 0.0f, d2 = 0.0f;
#pragma unroll 8
  for (int j = 0; j < OUT_DIM; ++j) {
    const float v = zr[j];
    d1 += v * attn_w[j];
    d2 += v * attn_w[OUT_DIM + j];
  }
  s1[n] = d1;
  s2[n] = d2;
}

// Monotone float<->uint encoding for atomicMax-based segment_max.
__device__ __forceinline__ unsigned enc_f32(float f) {
  unsigned b = __float_as_uint(f);
  return (b & 0x80000000u) ? ~b : (b | 0x80000000u);
}
__device__ __forceinline__ float dec_f32(unsigned e) {
  unsigned b = (e & 0x80000000u) ? (e & 0x7FFFFFFFu) : ~e;
  return __uint_as_float(b);
}

// ---------------------------------------------------------------------------
// Kernel 3: per-edge leaky-relu logit, store e, atomic segment-max into m_enc
// ---------------------------------------------------------------------------
__global__ __launch_bounds__(256) void k_edge_max(const int* __restrict__ src,
                                                  const int* __restrict__ dst,
                                                  const float* __restrict__ s1,
                                                  const float* __restrict__ s2,
                                                  float* __restrict__ ebuf,
                                                  unsigned* __restrict__ m_enc) {
  const int ei = blockIdx.x * blockDim.x + threadIdx.x;
  if (ei >= N_EDGES) return;
  const float logit = s1[src[ei]] + s2[dst[ei]];
  const float e = (logit >= 0.0f) ? logit : LEAK * logit;
  ebuf[ei] = e;
  atomicMax(&m_enc[dst[ei]], enc_f32(e));
}

// ---------------------------------------------------------------------------
// Kernel 4: ex = exp(e - m[dst]) (in place over ebuf), atomic segment-sum denom
// ---------------------------------------------------------------------------
__global__ __launch_bounds__(256) void k_edge_exp(const int* __restrict__ dst,
                                                  float* __restrict__ ebuf,
                                                  const unsigned* __restrict__ m_enc,
                                                  float* __restrict__ denom) {
  const int ei = blockIdx.x * blockDim.x + threadIdx.x;
  if (ei >= N_EDGES) return;
  const int d = dst[ei];
  const float ex = expf(ebuf[ei] - dec_f32(m_enc[d]));
  ebuf[ei] = ex;
  atomicAdd(&denom[d], ex);
}

// ---------------------------------------------------------------------------
// Kernel 5: h_out[dst] += (ex/denom[dst]) * z[src]; one wave per edge,
// each lane carries a float4 slice of the 128-wide row (L2-resident traffic).
// ---------------------------------------------------------------------------
__global__ __launch_bounds__(256) void k_scatter(const int* __restrict__ src,
                                                 const int* __restrict__ dst,
                                                 const float* __restrict__ z,
                                                 const float* __restrict__ ebuf,
                                                 const float* __restrict__ denom,
                                                 float* __restrict__ hacc) {
  const int ei = blockIdx.x * 8 + (threadIdx.x >> 5);
  if (ei >= N_EDGES) return;
  const int lane = threadIdx.x & 31;
  const int s = src[ei];
  const int d = dst[ei];
  const float dn = denom[d];
  const float alpha = ebuf[ei] / ((dn > 0.0f) ? dn : 1.0f);

  const v4f v = ((const v4f*)(z + (size_t)s * OUT_DIM))[lane];
  float* ho = hacc + (size_t)d * OUT_DIM + lane * 4;
  atomicAdd(&ho[0], alpha * v.x);
  atomicAdd(&ho[1], alpha * v.y);
  atomicAdd(&ho[2], alpha * v.z);
  atomicAdd(&ho[3], alpha * v.w);
}

// ---------------------------------------------------------------------------
// Kernel 6: in-place ELU-like epilogue: x>0 ? x : expm1(x)
// ---------------------------------------------------------------------------
__global__ __launch_bounds__(256) void k_final(float* __restrict__ out) {
  const int i = blockIdx.x * blockDim.x + threadIdx.x;
  if (i >= N_NODES * OUT_DIM) return;
  const float x = out[i];
  out[i] = (x > 0.0f) ? x : expm1f(x);
}

// ---------------------------------------------------------------------------
extern "C" void kernel_launch(void* const* d_in, const int* in_sizes, int n_in,
                              void* d_out, int out_size, void* d_ws, size_t ws_size,
                              hipStream_t stream) {
  const float* h      = (const float*)d_in[0];
  const int*   src    = (const int*)  d_in[1];
  const int*   dst    = (const int*)  d_in[2];
  const float* W_fc   = (const float*)d_in[3];
  const float* attn_w = (const float*)d_in[4];
  float*       out    = (float*)d_out;

  // Workspace layout (floats unless noted)
  float*    z     = (float*)d_ws;                         // 50000*128
  float*    s1    = z  + (size_t)N_NODES * OUT_DIM;       // 50000
  float*    s2    = s1 + N_NODES;                         // 50000
  float*    denom = s2 + N_NODES;                         // 50000
  unsigned* m_enc = (unsigned*)(denom + N_NODES);         // 50000 (uint)
  float*    ebuf  = (float*)(m_enc + N_NODES);            // 800000

  const int TPB = 256;

  k_init<<<(N_NODES * OUT_DIM + TPB - 1) / TPB, TPB, 0, stream>>>(out, denom, m_enc);

  k_gemm<<<N_NODES / 16, TPB, 0, stream>>>(h, W_fc, z);   // 3125 blocks x 8 waves

  k_scores<<<(N_NODES + TPB - 1) / TPB, TPB, 0, stream>>>(z, attn_w, s1, s2);

  k_edge_max<<<(N_EDGES + TPB - 1) / TPB, TPB, 0, stream>>>(src, dst, s1, s2, ebuf, m_enc);

  k_edge_exp<<<(N_EDGES + TPB - 1) / TPB, TPB, 0, stream>>>(dst, ebuf, m_enc, denom);

  k_scatter<<<(N_EDGES + 7) / 8, TPB, 0, stream>>>(src, dst, z, ebuf, denom, out);

  k_final<<<(N_NODES * OUT_DIM + TPB - 1) / TPB, TPB, 0, stream>>>(out);
}